// Similarity_Search_31009663877244
// MI455X (gfx1250) — compile-verified
//
#include <hip/hip_runtime.h>
#include <hip/hip_bf16.h>

// ---------------------------------------------------------------------------
// Similarity search: sims = desc(128x128) @ X^T(128x1M), top-10/query, vote.
// Phase 1: WMMA f32 16x16x4 GEMM; each wave owns TWO 16-col N-tiles so every
//          LDS A-fragment feeds two WMMAs; B loads are unconditional
//          nontemporal streams (no EXEC branching in the hot loop).
// Phase 2: per-query reduction of per-block top-10s + voting logic.
// Phase 3: pass-through of final_boxes.
// ---------------------------------------------------------------------------

typedef __attribute__((ext_vector_type(2))) float v2f;
typedef __attribute__((ext_vector_type(8))) float v8f;

#define D_Q        128          // queries (descriptors rows)
#define C_DIM      128          // feature dim
#define N_DB       1000000      // database rows
#define DB_STRIDE  129          // floats per db row (128 feats + place id)
#define TOPK       10
#define MIN_SIM    0.8f
#define A_PAD      132          // LDS row pitch for A (bank stagger)
#define TILE_N     256          // sims columns per block-tile (8 waves * 32)
#define S_PAD      (TILE_N + 1) // LDS row pitch for sims tile (conflict-free)

// ---------------------------------------------------------------------------
// Phase 1: each block owns `colsPerBlock` database rows (columns of sims).
// 8 waves/block; wave w computes the 128x32 sims sub-tile for columns
// [tile + w*32, tile + w*32 + 32) via V_WMMA_F32_16X16X4_F32.
// ---------------------------------------------------------------------------
__global__ void __launch_bounds__(256)
sims_topk_partial(const float* __restrict__ desc,
                  const float* __restrict__ db,
                  float* __restrict__ wsV, int* __restrict__ wsI,
                  int colsPerBlock)
{
    extern __shared__ float smem[];
    float* As = smem;                      // [128][A_PAD] descriptors
    float* Ss = smem + D_Q * A_PAD;        // [128][S_PAD] sims tile

    const int tid = threadIdx.x;

    // Stage descriptors into LDS (coalesced).
    for (int i = tid; i < D_Q * C_DIM; i += 256) {
        int r = i >> 7, c = i & 127;
        As[r * A_PAD + c] = desc[i];
    }

    const int wave  = tid >> 5;
    const int lane  = tid & 31;
    const int lhalf = lane >> 4;   // 0: K/M lanes 0-15, 1: lanes 16-31
    const int lmod  = lane & 15;

    // Per-query (owner threads tid<128) sorted top-10, descending.
    float topv[TOPK];
    int   topi[TOPK];
#pragma unroll
    for (int j = 0; j < TOPK; ++j) { topv[j] = -3.0e38f; topi[j] = -1; }

    const int colBase = blockIdx.x * colsPerBlock;
    __syncthreads();

    const int nTiles = colsPerBlock / TILE_N;
    for (int tile = 0; tile < nTiles; ++tile) {
        const int tileCol = colBase + tile * TILE_N;
        // This wave's two B columns for this lane. Out-of-range columns are
        // clamped to row 0: their sims get computed but are rejected by the
        // g < N_DB guard in the scan, so no predication is needed here.
        const int c0 = tileCol + wave * 32 + lmod;
        const int c1 = c0 + 16;
        const float* xrow0 = db + (size_t)(c0 < N_DB ? c0 : 0) * DB_STRIDE;
        const float* xrow1 = db + (size_t)(c1 < N_DB ? c1 : 0) * DB_STRIDE;

        v8f acc0[8] = {};   // 128 queries x cols [c0 tile]
        v8f acc1[8] = {};   // 128 queries x cols [c1 tile]

        for (int ks = 0; ks < 32; ++ks) {
            const int k = (ks << 2) + (lhalf << 1);   // K base for this lane half
            // B fragments: B[k][n]=X[col n][k]; VGPR0: K={0,2}, VGPR1: K={1,3}.
            // Streamed once -> nontemporal (TH=NT), keep L2 for hot data.
            v2f b0, b1;
            b0.x = __builtin_nontemporal_load(xrow0 + k);
            b0.y = __builtin_nontemporal_load(xrow0 + k + 1);
            b1.x = __builtin_nontemporal_load(xrow1 + k);
            b1.y = __builtin_nontemporal_load(xrow1 + k + 1);
#pragma unroll
            for (int mt = 0; mt < 8; ++mt) {
                // A fragment: lanes 0-15 M rows, VGPR0: K={0,2}, VGPR1: K={1,3}.
                v2f a;
                const int m = mt * 16 + lmod;
                a.x = As[m * A_PAD + k];
                a.y = As[m * A_PAD + k + 1];
                acc0[mt] = __builtin_amdgcn_wmma_f32_16x16x4_f32(
                    false, a, false, b0, (short)0, acc0[mt], false, false);
                acc1[mt] = __builtin_amdgcn_wmma_f32_16x16x4_f32(
                    false, a, false, b1, (short)0, acc1[mt], false, false);
            }
        }

        // Stage sims tile to LDS. (Barrier: previous tile's scan must finish.)
        __syncthreads();
#pragma unroll
        for (int mt = 0; mt < 8; ++mt) {
#pragma unroll
            for (int r = 0; r < 8; ++r) {
                // C/D layout: VGPR r -> M = r (lanes 0-15) / 8+r (lanes 16-31)
                const int m = mt * 16 + r + 8 * lhalf;
                const int c = wave * 32 + lmod;
                Ss[m * S_PAD + c]      = acc0[mt][r];
                Ss[m * S_PAD + c + 16] = acc1[mt][r];
            }
        }
        __syncthreads();

        // Owner threads scan their query row of the tile into the top-10.
        if (tid < D_Q) {
            const float* row = Ss + tid * S_PAD;
            for (int c = 0; c < TILE_N; ++c) {
                float v = row[c];
                int   g = tileCol + c;
                if (g < N_DB && v > topv[TOPK - 1]) {
                    int gi = g;
#pragma unroll
                    for (int j = 0; j < TOPK; ++j) {
                        if (v > topv[j]) {
                            float tv = topv[j]; topv[j] = v; v = tv;
                            int   ti = topi[j]; topi[j] = gi; gi = ti;
                        }
                    }
                }
            }
        }
    }

    if (tid < D_Q) {
        const size_t base = ((size_t)tid * gridDim.x + blockIdx.x) * TOPK;
#pragma unroll
        for (int j = 0; j < TOPK; ++j) {
            wsV[base + j] = topv[j];
            wsI[base + j] = topi[j];
        }
    }
}

// ---------------------------------------------------------------------------
// Phase 2: one block per query. Reduce nblk*10 candidates to global top-10,
// then apply the exact voting logic of the reference.
// ---------------------------------------------------------------------------
__global__ void __launch_bounds__(256)
topk_reduce_vote(const float* __restrict__ wsV, const int* __restrict__ wsI,
                 const float* __restrict__ db,
                 float* __restrict__ scoresOut, void* __restrict__ classesOut,
                 int cls64, int nblk)
{
    __shared__ float sv[256 * TOPK];
    __shared__ int   si[256 * TOPK];

    const int q   = blockIdx.x;
    const int tid = threadIdx.x;
    const int total = nblk * TOPK;

    float tv[TOPK]; int ti[TOPK];
#pragma unroll
    for (int j = 0; j < TOPK; ++j) { tv[j] = -3.0e38f; ti[j] = -1; }

    const float* V = wsV + (size_t)q * nblk * TOPK;
    const int*   I = wsI + (size_t)q * nblk * TOPK;

    for (int c = tid; c < total; c += 256) {
        float v = V[c];
        int   g = I[c];
        if (g >= 0 && v > tv[TOPK - 1]) {
            int gi = g;
#pragma unroll
            for (int j = 0; j < TOPK; ++j) {
                if (v > tv[j]) {
                    float t0 = tv[j]; tv[j] = v; v = t0;
                    int   t1 = ti[j]; ti[j] = gi; gi = t1;
                }
            }
        }
    }
#pragma unroll
    for (int j = 0; j < TOPK; ++j) { sv[tid * TOPK + j] = tv[j]; si[tid * TOPK + j] = ti[j]; }
    __syncthreads();

    if (tid == 0) {
        float bv[TOPK]; int bi[TOPK];
#pragma unroll
        for (int j = 0; j < TOPK; ++j) { bv[j] = -3.0e38f; bi[j] = -1; }
        for (int c = 0; c < 256 * TOPK; ++c) {
            float v = sv[c];
            int   g = si[c];
            if (g >= 0 && v > bv[TOPK - 1]) {
                int gi = g;
#pragma unroll
                for (int j = 0; j < TOPK; ++j) {
                    if (v > bv[j]) {
                        float t0 = bv[j]; bv[j] = v; v = t0;
                        int   t1 = bi[j]; bi[j] = gi; gi = t1;
                    }
                }
            }
        }

        // Voting (exact reference semantics).
        int  place[TOPK];
        bool mask[TOPK];
        int  nk = 0;
#pragma unroll
        for (int j = 0; j < TOPK; ++j) {
            place[j] = (bi[j] >= 0) ? (int)db[(size_t)bi[j] * DB_STRIDE + C_DIM] : -2;
            mask[j]  = (bi[j] >= 0) && (bv[j] >= MIN_SIM);
            nk += mask[j] ? 1 : 0;
        }
        int counts[TOPK], maxc = 0;
#pragma unroll
        for (int j = 0; j < TOPK; ++j) {
            int cnt = 0;
#pragma unroll
            for (int k = 0; k < TOPK; ++k)
                cnt += (mask[k] && place[k] == place[j]) ? 1 : 0;
            counts[j] = mask[j] ? cnt : 0;
            maxc = counts[j] > maxc ? counts[j] : maxc;
        }
        const int BIG = 1 << 30;
        int maj = BIG;
#pragma unroll
        for (int j = 0; j < TOPK; ++j)
            if (mask[j] && counts[j] == maxc && place[j] < maj) maj = place[j];

        const bool valid = (nk > 0);                       // MIN_VOTES == 0
        const int  cls   = valid ? maj : -1;
        const int  ref   = valid ? maj : -1;
        float best = -1.0e30f;
        bool  any  = false;
#pragma unroll
        for (int j = 0; j < TOPK; ++j) {
            bool match = (place[j] == ref);
            any = any || match;
            if (match && bv[j] > best) best = bv[j];
        }
        const float score = (valid && any) ? best : 0.0f;

        scoresOut[q] = score;
        if (cls64) ((long long*)classesOut)[q] = (long long)cls;
        else       ((float*)classesOut)[q]     = (float)cls;
    }
}

// ---------------------------------------------------------------------------
// Phase 3: final_boxes pass-through (512 floats).
// ---------------------------------------------------------------------------
__global__ void copy_boxes(const float* __restrict__ b, float* __restrict__ o)
{
    int i = blockIdx.x * 256 + threadIdx.x;
    if (i < D_Q * 4) o[i] = b[i];
}

// ---------------------------------------------------------------------------
// Host launcher.
// Inputs: 0=final_boxes f32[128*4], 1=final_scores f32[128],
//         2=final_classes i64[128], 3=descriptors f32[128*128],
//         4=places_db f32[1e6*129]
// Output: [boxes 512 f32][scores 128 f32][classes 128 (i64 if room else f32)]
// ---------------------------------------------------------------------------
extern "C" void kernel_launch(void* const* d_in, const int* in_sizes, int n_in,
                              void* d_out, int out_size, void* d_ws, size_t ws_size,
                              hipStream_t stream)
{
    const float* boxes = (const float*)d_in[0];
    const float* desc  = (const float*)d_in[3];
    const float* db    = (const float*)d_in[4];
    float* out = (float*)d_out;

    // Size phase-1 grid so per-block top-10s fit in workspace.
    size_t capBlocks = ws_size / ((size_t)D_Q * TOPK * (sizeof(float) + sizeof(int)));
    int nblk = (int)(capBlocks < 245 ? capBlocks : 245);
    if (nblk < 1) nblk = 1;
    if (nblk > 256) nblk = 256;                       // phase-2 LDS holds 256*10
    int cols = ((N_DB + nblk - 1) / nblk + TILE_N - 1) & ~(TILE_N - 1);
    nblk = (N_DB + cols - 1) / cols;

    float* wsV = (float*)d_ws;
    int*   wsI = (int*)((char*)d_ws + (size_t)D_Q * nblk * TOPK * sizeof(float));

    const size_t smemBytes = (size_t)(D_Q * A_PAD + D_Q * S_PAD) * sizeof(float);
    sims_topk_partial<<<nblk, 256, smemBytes, stream>>>(desc, db, wsV, wsI, cols);

    const int cls64 = (out_size >= 640 + 2 * D_Q) ? 1 : 0;   // room for i64 classes?
    void* clsOut = (void*)(out + 640);
    topk_reduce_vote<<<D_Q, 256, 0, stream>>>(wsV, wsI, db, out + 512, clsOut,
                                              cls64, nblk);

    copy_boxes<<<2, 256, 0, stream>>>(boxes, out);
}